// BatchNormalizationThroughTime1D_4209067950399
// MI455X (gfx1250) — compile-verified
//
#include <hip/hip_runtime.h>

// BNTT fused batchnorm-through-time, MI455X (gfx1250, wave32).
// Bandwidth-bound: ~390 MB HBM traffic -> ~17us floor at 23.3 TB/s.
// Pass 1 uses the CDNA5 async global->LDS data mover for the streaming
// reduction; pass 3 uses non-temporal float4 streams.

typedef float f32x4 __attribute__((ext_vector_type(4)));

#define B_   64
#define C_   256
#define T_   2048
#define NSEG 16                          // segments over B
#define ROWS_PER_SEG (B_ / NSEG)         // 4 batch rows per block
#define F4_PER_ROW   (T_ / 4)            // 512 float4 per (b,c) row
#define F4_PER_BLOCK (ROWS_PER_SEG * F4_PER_ROW)   // 2048
#define ITERS        (F4_PER_BLOCK / 256)          // 8
#define EPS_ 1e-4f

// ws layout (floats)
#define WS_SUM  0
#define WS_SQ   (C_ * NSEG)
#define WS_MEAN (2 * C_ * NSEG)
#define WS_INV  (2 * C_ * NSEG + C_)

// ---------------------------------------------------------------------------
// Pass 1: per-(channel, segment) partial sum / sumsq.
// Each wave double-buffers 32 x 16B tiles through LDS using
// global_load_async_to_lds_b128 (GVS: sgpr base + 32-bit vgpr byte offset),
// pipelined with s_wait_asynccnt. Wave-private staging => no barriers needed
// inside the streaming loop.
// ---------------------------------------------------------------------------
__global__ __launch_bounds__(256) void bntt_partial(const float* __restrict__ x,
                                                    float* __restrict__ ws) {
    const int c    = blockIdx.x;
    const int s    = blockIdx.y;
    const int tid  = threadIdx.x;
    const int lane = tid & 31;
    const int wave = tid >> 5;

    __shared__ f32x4 stage[8][2][32];   // [wave][buf][lane], 8 KB
    __shared__ float rsum[256];
    __shared__ float rsq[256];

    // LDS byte offsets of this lane's two staging slots (generic-pointer
    // low 32 bits == LDS offset on gfx1250).
    const unsigned lds0 = (unsigned)(unsigned long long)&stage[wave][0][lane];
    const unsigned lds1 = (unsigned)(unsigned long long)&stage[wave][1][lane];

    // Global byte offset of float4 #f of this block's tile.
    // f -> row r = f/512 within segment, j = f%512 float4 within row.
    auto goff = [&](int f) -> unsigned {
        const int r = f >> 9;
        const int j = f & 511;
        const int b = s * ROWS_PER_SEG + r;
        return (unsigned)((((b * C_ + c) * T_) + j * 4) * 4);
    };

    // Prime the pipeline: chunk 0 -> buffer 0.
    {
        const unsigned off = goff(tid);
        asm volatile("global_load_async_to_lds_b128 %0, %1, %2"
                     :: "v"(lds0), "v"(off), "s"(x) : "memory");
    }

    float accs = 0.0f, accq = 0.0f;
    for (int i = 0; i < ITERS; ++i) {
        if (i + 1 < ITERS) {
            const unsigned off = goff((i + 1) * 256 + tid);
            const unsigned dst = ((i + 1) & 1) ? lds1 : lds0;
            asm volatile("global_load_async_to_lds_b128 %0, %1, %2"
                         :: "v"(dst), "v"(off), "s"(x) : "memory");
            asm volatile("s_wait_asynccnt 0x1" ::: "memory"); // chunk i done
        } else {
            asm volatile("s_wait_asynccnt 0x0" ::: "memory"); // drain
        }
        const f32x4 v = stage[wave][i & 1][lane];   // ds_load_b128
        accs += (v.x + v.y) + (v.z + v.w);
        accq += (v.x * v.x + v.y * v.y) + (v.z * v.z + v.w * v.w);
    }

    rsum[tid] = accs;
    rsq[tid]  = accq;
    __syncthreads();
    // Fixed-order tree reduction => deterministic result every call.
    for (int step = 128; step > 0; step >>= 1) {
        if (tid < step) {
            rsum[tid] += rsum[tid + step];
            rsq[tid]  += rsq[tid + step];
        }
        __syncthreads();
    }
    if (tid == 0) {
        ws[WS_SUM + c * NSEG + s] = rsum[0];
        ws[WS_SQ  + c * NSEG + s] = rsq[0];
    }
}

// ---------------------------------------------------------------------------
// Pass 2: fold 16 partials per channel -> mean, invstd.
// ---------------------------------------------------------------------------
__global__ __launch_bounds__(256) void bntt_finalize(float* __restrict__ ws) {
    const int c = threadIdx.x;   // one thread per channel, single block
    float sum = 0.0f, sq = 0.0f;
    #pragma unroll
    for (int s = 0; s < NSEG; ++s) {
        sum += ws[WS_SUM + c * NSEG + s];
        sq  += ws[WS_SQ  + c * NSEG + s];
    }
    const float invN = 1.0f / (float)(B_ * T_);
    const float mean = sum * invN;
    const float var  = sq * invN - mean * mean;
    ws[WS_MEAN + c] = mean;
    ws[WS_INV  + c] = rsqrtf(var + EPS_);
}

// ---------------------------------------------------------------------------
// Pass 3: out = x*scale + shift with scale = gamma[t,c]*inv,
// shift = beta[t,c] - mean*scale. Block = (channel, 1024-wide t chunk);
// gamma/beta fetched once per block (L2-resident), x/out streamed NT float4.
// ---------------------------------------------------------------------------
__global__ __launch_bounds__(256) void bntt_apply(const float* __restrict__ x,
                                                  const float* __restrict__ gamma,
                                                  const float* __restrict__ beta,
                                                  const float* __restrict__ ws,
                                                  float* __restrict__ out) {
    const int c  = blockIdx.x;
    const int t0 = (blockIdx.y * 256 + threadIdx.x) * 4;

    const float mean = ws[WS_MEAN + c];
    const float inv  = ws[WS_INV  + c];

    f32x4 scale, shift;
    #pragma unroll
    for (int k = 0; k < 4; ++k) {
        const float g  = gamma[(size_t)(t0 + k) * C_ + c];
        const float bt = beta [(size_t)(t0 + k) * C_ + c];
        const float sc = g * inv;
        scale[k] = sc;
        shift[k] = fmaf(-mean, sc, bt);
    }

    #pragma unroll 4
    for (int b = 0; b < B_; ++b) {
        const size_t base = (size_t)(b * C_ + c) * T_ + t0;
        const f32x4 xv = __builtin_nontemporal_load((const f32x4*)(x + base));
        f32x4 ov;
        ov.x = fmaf(xv.x, scale.x, shift.x);
        ov.y = fmaf(xv.y, scale.y, shift.y);
        ov.z = fmaf(xv.z, scale.z, shift.z);
        ov.w = fmaf(xv.w, scale.w, shift.w);
        __builtin_nontemporal_store(ov, (f32x4*)(out + base));
    }
}

// ---------------------------------------------------------------------------
extern "C" void kernel_launch(void* const* d_in, const int* in_sizes, int n_in,
                              void* d_out, int out_size, void* d_ws, size_t ws_size,
                              hipStream_t stream) {
    const float* x     = (const float*)d_in[0];   // [B, C, T]
    const float* gamma = (const float*)d_in[1];   // [T, C]
    const float* beta  = (const float*)d_in[2];   // [T, C]
    float*       out   = (float*)d_out;           // [B, C, T]
    float*       ws    = (float*)d_ws;            // 34 KB used

    bntt_partial <<<dim3(C_, NSEG),        256, 0, stream>>>(x, ws);
    bntt_finalize<<<1,                     256, 0, stream>>>(ws);
    bntt_apply   <<<dim3(C_, T_ / 1024),   256, 0, stream>>>(x, gamma, beta, ws, out);
}